// Burden_29145648070955
// MI455X (gfx1250) — compile-verified
//
#include <hip/hip_runtime.h>

typedef float v2f __attribute__((ext_vector_type(2)));
typedef float v8f __attribute__((ext_vector_type(8)));

#define X_DIM   1024
#define BATCH   65536
#define N_REC   21          // sigma_1 -> sigma_22
#define WAVES_PER_BLOCK 8
#define ROWS_PER_WAVE   16
#define ROWS_PER_BLOCK  (WAVES_PER_BLOCK * ROWS_PER_WAVE)   // 128

// out[r] = sigma_22 where sigma_1 = X[r]@w + b,
//          sigma_{k+1} = sigma_1 + 0.5*wn2*nabla_f(sigma_k),
//          nabla_f(s) = 0.5*(s+1)*rsqrt((s+1)^2+1), wn2 = ||w||^2.
__global__ __launch_bounds__(256) void ccp_score_kernel(
    const float* __restrict__ X,
    const float* __restrict__ w,
    const float* __restrict__ b,
    float* __restrict__ out)
{
    __shared__ float lds_w[X_DIM];                          // 4 KB staged w
    __shared__ float lds_dot[WAVES_PER_BLOCK * ROWS_PER_WAVE];
    __shared__ float lds_red[WAVES_PER_BLOCK];

    const int tid  = threadIdx.x;
    const int lane = tid & 31;
    const int wave = tid >> 5;

    // Stage w into LDS (256 threads x float4, coalesced) and compute ||w||^2
    // once in the prologue via wave shfl-reduction + per-wave LDS partials.
    {
        float4 v = ((const float4*)w)[tid];
        ((float4*)lds_w)[tid] = v;
        float p = v.x * v.x + v.y * v.y + v.z * v.z + v.w * v.w;
#pragma unroll
        for (int off = 16; off >= 1; off >>= 1)
            p += __shfl_xor(p, off, 32);
        if (lane == 0) lds_red[wave] = p;
    }
    __syncthreads();

    const int rowBase = blockIdx.x * ROWS_PER_BLOCK + wave * ROWS_PER_WAVE;
    const int myRow   = rowBase + (lane & 15);
    const int ksub    = (lane >> 4) * 2;   // lanes 0-15 -> K+{0,1}; lanes 16-31 -> K+{2,3}

    const float* __restrict__ xrow = X + (size_t)myRow * X_DIM + ksub;
    const float* __restrict__ wl   = lds_w + ksub;

    v8f acc0 = {};
    v8f acc1 = {};

    // Hot loop: A from global (b64/lane), B = broadcast-w from LDS (b64/lane),
    // two independent WMMA accumulation chains. Nothing else.
#pragma unroll 2
    for (int k0 = 0; k0 < X_DIM; k0 += 8) {
        v2f a0 = *(const v2f*)(xrow + k0);
        v2f b0 = *(const v2f*)(wl   + k0);
        v2f a1 = *(const v2f*)(xrow + k0 + 4);
        v2f b1 = *(const v2f*)(wl   + k0 + 4);
        acc0 = __builtin_amdgcn_wmma_f32_16x16x4_f32(
            /*neg_a=*/false, a0, /*neg_b=*/false, b0,
            /*c_mod=*/(short)0, acc0, /*reuse_a=*/false, /*reuse_b=*/false);
        acc1 = __builtin_amdgcn_wmma_f32_16x16x4_f32(
            /*neg_a=*/false, a1, /*neg_b=*/false, b1,
            /*c_mod=*/(short)0, acc1, /*reuse_a=*/false, /*reuse_b=*/false);
    }

    v8f acc = acc0 + acc1;

    // C layout (16x16 f32): lane 0 VGPR v holds row M=v (N=0);
    // lane 16 VGPR v holds row M=v+8 (N=0). All N columns identical.
    if ((lane & 15) == 0) {
        float* dst = lds_dot + wave * ROWS_PER_WAVE + (lane >> 4) * 8;
#pragma unroll
        for (int i = 0; i < 8; ++i) dst[i] = acc[i];
    }
    __syncthreads();

    if (lane < 16) {
        float wn2 = 0.0f;
#pragma unroll
        for (int i = 0; i < WAVES_PER_BLOCK; ++i) wn2 += lds_red[i];

        const float dot = lds_dot[wave * ROWS_PER_WAVE + lane];
        const float s0  = dot + b[0];
        const float hw2 = 0.5f * wn2;
        float s = s0;
#pragma unroll
        for (int t = 0; t < N_REC; ++t) {
            float z   = s + 1.0f;
            float nab = 0.5f * z * rsqrtf(__builtin_fmaf(z, z, 1.0f));
            s = __builtin_fmaf(hw2, nab, s0);
        }
        out[rowBase + lane] = s;
    }
}

extern "C" void kernel_launch(void* const* d_in, const int* in_sizes, int n_in,
                              void* d_out, int out_size, void* d_ws, size_t ws_size,
                              hipStream_t stream) {
    const float* X = (const float*)d_in[0];   // [65536, 1024] f32
    const float* w = (const float*)d_in[1];   // [1024] f32
    const float* b = (const float*)d_in[2];   // [1] f32
    float* out = (float*)d_out;               // [65536] f32

    const int blocks = BATCH / ROWS_PER_BLOCK;   // 512
    ccp_score_kernel<<<blocks, 256, 0, stream>>>(X, w, b, out);
}